// PQHead_35502199669511
// MI455X (gfx1250) — compile-verified
//
#include <hip/hip_runtime.h>
#include <stdint.h>

typedef __attribute__((ext_vector_type(2))) float v2f;
typedef __attribute__((ext_vector_type(8))) float v8f;

#define BDIM 768
#define MSUB 128
#define DSUB 6
#define CBROW (32 * DSUB)   // 192 floats per subspace
#define ROWS  32            // two 16-row tiles per wave

// XOR lane swap via ds_swizzle immediate (group-of-32): and=0x1f, or=0, xor=s
#define SWZ_XOR(v, s) __builtin_amdgcn_ds_swizzle((v), (0x1f | ((s) << 10)))

// strict-greater tournament step: keeps the earlier (lower-code) entry on ties
__device__ __forceinline__ void vimax(float& v, int& i, float v2, int i2) {
    bool c = v2 > v;
    v = c ? v2 : v;
    i = c ? i2 : i;
}

__global__ __launch_bounds__(256) void pq_head_kernel(const float* __restrict__ x,
                                                      const float* __restrict__ cb,
                                                      float* __restrict__ out) {
    const int lane = threadIdx.x & 31;
    const int wave = threadIdx.x >> 5;
    const int b0   = blockIdx.x * ROWS;
    const int m    = blockIdx.y * 8 + wave;      // one subspace per wave

    const int n    = lane & 15;
    const int half = lane >> 4;
    const float* cbm = cb + (size_t)m * CBROW;

    // ---- A = codebook (16 codes x K), tile-invariant.
    // A layout (16x4 f32): lanes 0-15 hold K={0,1}, lanes 16-31 K={2,3}, M=lane&15.
    const float* c0 = cbm + n * DSUB;            // codes 0..15
    const float* c1 = cbm + (n + 16) * DSUB;     // codes 16..31
    v2f a00 = *(const v2f*)(c0 + 2 * half);      // K {0,1}/{2,3}
    v2f a10 = *(const v2f*)(c1 + 2 * half);
    v2f a01 = *(const v2f*)(c0 + 4);             // K {4,5}; K {6,7} pad on high half
    v2f a11 = *(const v2f*)(c1 + 4);
    if (half) { a01.x = 0.0f; a01.y = 0.0f; a11.x = 0.0f; a11.y = 0.0f; }

    #pragma unroll
    for (int t = 0; t < 2; ++t) {
        const int rb = b0 + t * 16;

        // ---- B = x (K x 16 rows): B[K][row] = x[rb+row][m*6+K]
        // B layout (4x16 f32): vgpr0 = K even (0 / 2 by half), vgpr1 = K odd, N=lane&15.
        const float* xrow = x + (size_t)(rb + n) * BDIM + m * DSUB;
        v2f bx0 = *(const v2f*)(xrow + 2 * half);
        v2f bx1 = *(const v2f*)(xrow + 4);       // K {4,5}; pad {6,7} on high half
        if (half) { bx1.x = 0.0f; bx1.y = 0.0f; }

        // ---- dotsT = A x B, exact fp32, K padded 6->8.
        // C layout: lane L<16 -> row L, reg r -> code half*8+r (acc0) / 16+half*8+r (acc1)
        v8f acc0 = {}; v8f acc1 = {};
        acc0 = __builtin_amdgcn_wmma_f32_16x16x4_f32(false, a00, false, bx0, (short)0, acc0, false, false);
        acc0 = __builtin_amdgcn_wmma_f32_16x16x4_f32(false, a01, false, bx1, (short)0, acc0, false, false);
        acc1 = __builtin_amdgcn_wmma_f32_16x16x4_f32(false, a10, false, bx0, (short)0, acc1, false, false);
        acc1 = __builtin_amdgcn_wmma_f32_16x16x4_f32(false, a11, false, bx1, (short)0, acc1, false, false);

        // ---- in-lane argmax over this lane's 16 codes (r-order = ascending code)
        float v[16]; int ix[16];
        #pragma unroll
        for (int r = 0; r < 8; ++r) { v[r] = acc0[r]; ix[r] = r; v[r + 8] = acc1[r]; ix[r + 8] = r + 8; }
        #pragma unroll
        for (int stride = 1; stride < 16; stride <<= 1)
            #pragma unroll
            for (int r = 0; r < 16; r += 2 * stride)
                vimax(v[r], ix[r], v[r + stride], ix[r + stride]);
        float bv = v[0];
        int   r  = ix[0];
        int code = (r & 7) + (half << 3) + ((r & 8) << 1);   // reg index -> global code

        // ---- resolve with partner lane (other 16 codes of this row), tie -> lower code
        float ov = __int_as_float(SWZ_XOR(__float_as_int(bv), 16));
        int   oc = SWZ_XOR(code, 16);
        bool take = (ov > bv) || ((ov == bv) && (oc < code));
        code = take ? oc : code;

        // ---- emit: row = lane&15; lanes 0-15 write floats 0-2, lanes 16-31 floats 3-5
        const int tri = half * 3;
        const float* src = cbm + code * DSUB + tri;                      // L2-hot gather
        float* dst = out + (size_t)(rb + n) * BDIM + m * DSUB + tri;
        dst[0] = src[0]; dst[1] = src[1]; dst[2] = src[2];
    }
}

extern "C" void kernel_launch(void* const* d_in, const int* in_sizes, int n_in,
                              void* d_out, int out_size, void* d_ws, size_t ws_size,
                              hipStream_t stream) {
    const float* x  = (const float*)d_in[0];   // (32768, 768) f32
    const float* cb = (const float*)d_in[1];   // (128, 32, 6) f32
    float* out = (float*)d_out;                // (32768, 768) f32
    dim3 grid(32768 / ROWS, MSUB / 8);
    pq_head_kernel<<<grid, dim3(256), 0, stream>>>(x, cb, out);
}